// GNNRefiner_240518168613
// MI455X (gfx1250) — compile-verified
//
#include <hip/hip_runtime.h>
#include <hip/hip_bf16.h>
#include <math.h>

typedef __attribute__((ext_vector_type(2))) float v2f;
typedef __attribute__((ext_vector_type(8))) float v8f;

#define N_NODES 28
#define NPAD 32
#define H 64
#define LDSW 36   // padded LDS row stride (floats) for the per-wave T tile

// workspace layout (float offsets)
#define WS_AHAT  0      // 32x32 padded dense normalized adjacency, Ahat[n*32+k]
#define WS_ZS    1024   // 64 sorted breakpoints
#define WS_SLOPE 1088   // 65 interval slopes
#define WS_INTER 1153   // 65 interval intercepts

// ---------------------------------------------------------------------------
// Prep: build dense Ahat (32x32, zero padded) and piecewise-linear tables for
//   f(v) = sum_o W2[o] * relu(W1[o]*v + b1[o])
// Single block; fully parallel (rank sort + Hillis-Steele scan + tree reduce).
// ---------------------------------------------------------------------------
__global__ __launch_bounds__(256)
void gcn_prep_kernel(const int* __restrict__ ei, int E,
                     const float* __restrict__ W1,
                     const float* __restrict__ b1,
                     const float* __restrict__ W2,
                     float* __restrict__ ws) {
  __shared__ float Alds[NPAD * NPAD];
  __shared__ float deg[NPAD];
  __shared__ float dinv[NPAD];
  __shared__ float zraw[H], dsl[H], din[H];
  __shared__ float zs[H], dss[H], dis[H];
  __shared__ float scs[H], sci[H];   // scan buffers
  __shared__ float bs[H], bi[H];     // base-sum reduction buffers
  int tid = threadIdx.x;

  for (int i = tid; i < NPAD * NPAD; i += blockDim.x) Alds[i] = 0.f;
  if (tid < NPAD) deg[tid] = (tid < N_NODES) ? 1.0f : 0.0f;  // self-loop deg
  __syncthreads();

  for (int e = tid; e < E; e += blockDim.x)
    atomicAdd(&deg[ei[E + e]], 1.0f);
  __syncthreads();

  if (tid < NPAD)
    dinv[tid] = (tid < N_NODES && deg[tid] > 0.f) ? rsqrtf(deg[tid]) : 0.f;
  __syncthreads();

  for (int e = tid; e < E; e += blockDim.x) {
    int s = ei[e], d = ei[E + e];
    atomicAdd(&Alds[d * NPAD + s], dinv[s] * dinv[d]);
  }
  if (tid < N_NODES)
    atomicAdd(&Alds[tid * NPAD + tid], dinv[tid] * dinv[tid]);  // self-loops
  __syncthreads();

  for (int i = tid; i < NPAD * NPAD; i += blockDim.x) ws[WS_AHAT + i] = Alds[i];

  // --- per-term breakpoint / delta / base contribution ---
  if (tid < H) {
    float w1 = W1[tid], bb = b1[tid], w2 = W2[tid];
    float z, dslope, dinter, bslope = 0.f, binter = 0.f;
    if (w1 != 0.f) {
      z = -bb / w1;
      if (w1 > 0.f) { dslope =  w1 * w2; dinter =  bb * w2; }  // turns ON as v grows
      else          { dslope = -w1 * w2; dinter = -bb * w2;    // turns OFF as v grows
                      bslope =  w1 * w2; binter =  bb * w2; }  // active at v=-inf
    } else {
      z = 3.0e38f; dslope = 0.f; dinter = 0.f;                 // never toggles
      if (bb > 0.f) binter = bb * w2;                          // constant ON term
    }
    zraw[tid] = z; dsl[tid] = dslope; din[tid] = dinter;
    bs[tid] = bslope; bi[tid] = binter;
  }
  __syncthreads();

  if (tid < H) {  // parallel rank sort of 64 breakpoints
    float z = zraw[tid];
    int rank = 0;
    for (int j = 0; j < H; ++j) {
      float zj = zraw[j];
      rank += (zj < z || (zj == z && j < tid)) ? 1 : 0;
    }
    zs[rank] = z; dss[rank] = dsl[tid]; dis[rank] = din[tid];
  }
  __syncthreads();

  // --- inclusive Hillis-Steele scan of sorted deltas (6 steps) ---
  if (tid < H) { scs[tid] = dss[tid]; sci[tid] = dis[tid]; }
  __syncthreads();
  for (int off = 1; off < H; off <<= 1) {
    float vs = 0.f, vi = 0.f;
    if (tid < H && tid >= off) { vs = scs[tid - off]; vi = sci[tid - off]; }
    __syncthreads();
    if (tid < H) { scs[tid] += vs; sci[tid] += vi; }
    __syncthreads();
  }

  // --- tree reduction for base slope/intercept at v = -inf ---
  for (int off = H / 2; off >= 1; off >>= 1) {
    if (tid < off) { bs[tid] += bs[tid + off]; bi[tid] += bi[tid + off]; }
    __syncthreads();
  }

  if (tid < H) {
    ws[WS_ZS + tid]        = zs[tid];
    ws[WS_SLOPE + tid + 1] = bs[0] + scs[tid];
    ws[WS_INTER + tid + 1] = bi[0] + sci[tid];
  }
  if (tid == 0) { ws[WS_SLOPE] = bs[0]; ws[WS_INTER] = bi[0]; }
}

// ---------------------------------------------------------------------------
// Main: out = Ahat * f(Ahat * x^T) + b2, batched as two WMMA GEMMs.
// 256 threads = 8 wave32; each wave owns 16 batch rows (16x32 tiles).
// V_WMMA_F32_16X16X4_F32, K-loop of 7 (K=28).
// ---------------------------------------------------------------------------
__global__ __launch_bounds__(256)
void gcn_wmma_kernel(const float* __restrict__ X,
                     const float* __restrict__ ws,
                     const float* __restrict__ b2p,
                     float* __restrict__ out, int B) {
  __shared__ float zsl[128];      // sorted breakpoints padded to 128 with +inf
  __shared__ float sll[H + 1];
  __shared__ float inl[H + 1];
  __shared__ float tt[8 * 16 * LDSW];  // per-wave 16x36 intermediate tiles

  int tid = threadIdx.x;
  for (int i = tid; i < 128; i += 256)
    zsl[i] = (i < H) ? ws[WS_ZS + i] : 3.402823466e38f;
  for (int i = tid; i < H + 1; i += 256) {
    sll[i] = ws[WS_SLOPE + i];
    inl[i] = ws[WS_INTER + i];
  }
  __syncthreads();

  int lane = tid & 31;
  int wave = tid >> 5;
  int hi   = lane >> 4;   // 0/1 : selects K sub-pair in A/B frags
  int lx   = lane & 15;   // M (A) or N (B/C/D) index
  int row0 = (blockIdx.x * 8 + wave) * 16;

  // --- A fragments: 16 batch rows of X, K = 0..27 (7 frags of K=4) ---
  // 16x4 f32 A layout: lanes 0-15 hold M=lane, K={0,1}; lanes 16-31 K={2,3}
  int rsafe = row0 + lx; if (rsafe >= B) rsafe = B - 1;
  v2f a[7];
#pragma unroll
  for (int k4 = 0; k4 < 7; ++k4)
    a[k4] = *(const v2f*)(X + (size_t)rsafe * N_NODES + k4 * 4 + 2 * hi);

  // --- B fragments: Ahat^T operand, identical for both GEMM stages ---
  // 4x16 f32 B layout mirrors A: lanes 0-15 hold N=lane, K={0,1}; 16-31 K={2,3}
  v2f b[2][7];
#pragma unroll
  for (int t = 0; t < 2; ++t)
#pragma unroll
    for (int k4 = 0; k4 < 7; ++k4)
      b[t][k4] = *(const v2f*)(ws + WS_AHAT + (lx + 16 * t) * NPAD + k4 * 4 + 2 * hi);

  // --- GEMM 1: S = X * Ahat^T  (16 x 32 tile) ---
  v8f c0 = {}; v8f c1 = {};
#pragma unroll
  for (int k4 = 0; k4 < 7; ++k4)
    c0 = __builtin_amdgcn_wmma_f32_16x16x4_f32(false, a[k4], false, b[0][k4],
                                               (short)0, c0, false, false);
#pragma unroll
  for (int k4 = 0; k4 < 7; ++k4)
    c1 = __builtin_amdgcn_wmma_f32_16x16x4_f32(false, a[k4], false, b[1][k4],
                                               (short)0, c1, false, false);

  // --- elementwise f via 7-step branchless lower_bound + 1 FMA; stash T in LDS ---
  float* tw = tt + wave * 16 * LDSW;
#pragma unroll
  for (int t = 0; t < 2; ++t) {
#pragma unroll
    for (int r = 0; r < 8; ++r) {
      float v = (t == 0) ? c0[r] : c1[r];
      int i = 0;
      i += (zsl[i + 63] < v) ? 64 : 0;
      i += (zsl[i + 31] < v) ? 32 : 0;
      i += (zsl[i + 15] < v) ? 16 : 0;
      i += (zsl[i +  7] < v) ?  8 : 0;
      i += (zsl[i +  3] < v) ?  4 : 0;
      i += (zsl[i +  1] < v) ?  2 : 0;
      i += (zsl[i     ] < v) ?  1 : 0;   // i = #breakpoints < v, in [0,64]
      float fv = fmaf(sll[i], v, inl[i]);
      int n = lx + 16 * t;
      if (n >= N_NODES) fv = 0.f;        // zero pad columns 28..31 for GEMM 2
      // C/D layout: lanes 0-15 -> M=r, lanes 16-31 -> M=r+8; N = lx+16t
      tw[(r + 8 * hi) * LDSW + n] = fv;
    }
  }
  // wave-synchronous LDS: same-wave DS ops are in order, no barrier needed

  // --- reload T in A-fragment layout ---
  v2f a2[7];
#pragma unroll
  for (int k4 = 0; k4 < 7; ++k4)
    a2[k4] = *(const v2f*)(tw + lx * LDSW + k4 * 4 + 2 * hi);

  // --- GEMM 2: Out = T * Ahat^T (same B frags; T cols 28..31 are zero) ---
  v8f o0 = {}; v8f o1 = {};
#pragma unroll
  for (int k4 = 0; k4 < 7; ++k4)
    o0 = __builtin_amdgcn_wmma_f32_16x16x4_f32(false, a2[k4], false, b[0][k4],
                                               (short)0, o0, false, false);
#pragma unroll
  for (int k4 = 0; k4 < 7; ++k4)
    o1 = __builtin_amdgcn_wmma_f32_16x16x4_f32(false, a2[k4], false, b[1][k4],
                                               (short)0, o1, false, false);

  float bias2 = b2p[0];
#pragma unroll
  for (int t = 0; t < 2; ++t) {
#pragma unroll
    for (int r = 0; r < 8; ++r) {
      int d = lx + 16 * t;
      int rr = row0 + r + 8 * hi;
      float val = ((t == 0) ? o0[r] : o1[r]) + bias2;
      if (d < N_NODES && rr < B)
        out[(size_t)rr * N_NODES + d] = val;
    }
  }
}

extern "C" void kernel_launch(void* const* d_in, const int* in_sizes, int n_in,
                              void* d_out, int out_size, void* d_ws, size_t ws_size,
                              hipStream_t stream) {
  const float* X  = (const float*)d_in[0];
  const int*   ei = (const int*)  d_in[1];   // edge_index, [2][E]
  const float* W1 = (const float*)d_in[2];
  const float* b1 = (const float*)d_in[3];
  const float* W2 = (const float*)d_in[4];
  const float* b2 = (const float*)d_in[5];
  float* ws = (float*)d_ws;
  int E = in_sizes[1] / 2;
  int B = in_sizes[0] / N_NODES;

  gcn_prep_kernel<<<1, 256, 0, stream>>>(ei, E, W1, b1, W2, ws);
  int blocks = (B + 127) / 128;  // 8 waves/block * 16 rows/wave
  gcn_wmma_kernel<<<blocks, 256, 0, stream>>>(X, ws, b2, (float*)d_out, B);
}